// SelfAttention2dBlock_14164802142380
// MI455X (gfx1250) — compile-verified
//
#include <hip/hip_runtime.h>
#include <math.h>

typedef float v2f __attribute__((ext_vector_type(2)));
typedef float v8f __attribute__((ext_vector_type(8)));

#define DIMS    64
#define BATCH   2
#define SPATIAL 1024            // H*W = 16*64
#define PLANE   (DIMS * SPATIAL) // one batch's (C,S) plane = 65536 floats

// ---------------------------------------------------------------------------
// One wave computes a 16x16 tile of  Y = W (64x64) @ X (64x1024) + bias,
// optionally + residual, via V_WMMA_F32_16X16X4_F32 (full fp32 precision).
//
// Layouts per CDNA5 ISA 7.12.2 (wave32):
//   A 16x4 f32 (2 VGPRs): lanes 0-15 -> M=lane, K={kk, kk+1};
//                          lanes 16-31 -> M=lane-16, K={kk+2, kk+3}
//   B 4x16 f32 (2 VGPRs): lanes 0-15 -> N=lane, rows K={kk, kk+1};
//                          lanes 16-31 -> N=lane-16, rows K={kk+2, kk+3}
//   C/D 16x16 f32 (8 VGPRs): VGPR r -> row m0 + r + 8*(lane>=16), col lane&15
// ---------------------------------------------------------------------------
__device__ __forceinline__ void wmma_gemm_tile_64x64(
    const float* __restrict__ W,     // 64x64 row-major weights
    const float* __restrict__ X,     // 64x1024 row-major activations (one batch)
    const float* __restrict__ bias,  // 64
    float* __restrict__ Y,           // 64x1024 output (one batch)
    const float* __restrict__ resid, // optional residual, same layout as Y
    int m0, int n0, int lane)
{
  const int half = lane >> 4;   // 0 for lanes 0-15, 1 for lanes 16-31
  const int l15  = lane & 15;
  const int arow = m0 + l15;    // A-matrix row index held by this lane
  const int klo  = half * 2;    // K sub-offset held by this lane
  const int bcol = n0 + l15;    // B/C/D column held by this lane

  v8f acc = {};
#pragma unroll
  for (int kk = 0; kk < DIMS; kk += 4) {
    v2f a, b;
    a.x = W[arow * DIMS + (kk + klo)];
    a.y = W[arow * DIMS + (kk + klo + 1)];
    b.x = X[(kk + klo) * SPATIAL + bcol];
    b.y = X[(kk + klo + 1) * SPATIAL + bcol];
    // 8-arg form: (neg_a, A, neg_b, B, c_mod, C, reuse_a, reuse_b)
    acc = __builtin_amdgcn_wmma_f32_16x16x4_f32(
        false, a, false, b, (short)0, acc, false, false);
  }

#pragma unroll
  for (int r = 0; r < 8; ++r) {
    const int row = m0 + r + 8 * half;
    const int idx = row * SPATIAL + bcol;
    float val = acc[r] + bias[row];
    if (resid) val += resid[idx];
    Y[idx] = val;
  }
}

// ---------------------------------------------------------------------------
// Fused Q/K/V projection. grid = (S/16, 64/16, 3*BATCH), block = 32 (1 wave).
// blockIdx.z: bit0 = batch, bits[2:1] = which projection (0=q,1=k,2=v).
// ---------------------------------------------------------------------------
__global__ __launch_bounds__(32)
void qkv_proj_kernel(const float* __restrict__ x,
                     const float* __restrict__ Wq, const float* __restrict__ bq,
                     const float* __restrict__ Wk, const float* __restrict__ bk,
                     const float* __restrict__ Wv, const float* __restrict__ bv,
                     float* __restrict__ q, float* __restrict__ k,
                     float* __restrict__ v)
{
  const int lane = threadIdx.x;
  const int n0 = blockIdx.x * 16;
  const int m0 = blockIdx.y * 16;
  const int b  = blockIdx.z & 1;
  const int p  = blockIdx.z >> 1;

  const float* W    = (p == 0) ? Wq : (p == 1) ? Wk : Wv;
  const float* bias = (p == 0) ? bq : (p == 1) ? bk : bv;
  float*       Y    = (p == 0) ? q  : (p == 1) ? k  : v;

  wmma_gemm_tile_64x64(W, x + b * PLANE, bias, Y + b * PLANE,
                       nullptr, m0, n0, lane);
}

// ---------------------------------------------------------------------------
// Rank-1 attention per head. scores[s,t] = q[s]*k[t]/8 is an outer product,
// so the softmax row-max is exactly q_s*kmax (q_s>=0) or q_s*kmin (q_s<0):
// one-pass softmax, no SxS score matrix ever materialized.
// grid = 128 heads, block = 256 threads; k,v cached in LDS (8 KB / 320 KB).
// ---------------------------------------------------------------------------
__global__ __launch_bounds__(256)
void attn_rank1_kernel(const float* __restrict__ q,
                       const float* __restrict__ k,
                       const float* __restrict__ v,
                       float* __restrict__ o)
{
  __shared__ float ks[SPATIAL];
  __shared__ float vs[SPATIAL];
  __shared__ float smax[256];
  __shared__ float smin[256];

  const int tid  = threadIdx.x;
  const int base = blockIdx.x * SPATIAL;  // head bg = b*64+ch is contiguous

  float lmax = -INFINITY, lmin = INFINITY;
  for (int i = tid; i < SPATIAL; i += 256) {
    const float kv = k[base + i];
    ks[i] = kv;
    vs[i] = v[base + i];
    lmax = fmaxf(lmax, kv);
    lmin = fminf(lmin, kv);
  }
  smax[tid] = lmax;
  smin[tid] = lmin;
  __syncthreads();
#pragma unroll
  for (int stride = 128; stride > 0; stride >>= 1) {
    if (tid < stride) {
      smax[tid] = fmaxf(smax[tid], smax[tid + stride]);
      smin[tid] = fminf(smin[tid], smin[tid + stride]);
    }
    __syncthreads();
  }
  const float kmax = smax[0];
  const float kmin = smin[0];

#pragma unroll
  for (int i = 0; i < SPATIAL / 256; ++i) {
    const int s = tid + 256 * i;
    const float qs = q[base + s] * 0.125f;             // 1/sqrt(DIMS)
    const float m  = (qs >= 0.0f) ? qs * kmax : qs * kmin;
    float den = 0.0f, num = 0.0f;
#pragma unroll 4
    for (int t = 0; t < SPATIAL; ++t) {
      const float e = __expf(qs * ks[t] - m);          // v_exp_f32 hot loop
      den += e;
      num += e * vs[t];
    }
    o[base + s] = num / den;
  }
}

// ---------------------------------------------------------------------------
// Output projection + bias + residual. grid = (S/16, 4, BATCH), block = 32.
// ---------------------------------------------------------------------------
__global__ __launch_bounds__(32)
void out_proj_kernel(const float* __restrict__ oatt,
                     const float* __restrict__ Wo, const float* __restrict__ bo,
                     const float* __restrict__ x, float* __restrict__ out)
{
  const int lane = threadIdx.x;
  const int n0 = blockIdx.x * 16;
  const int m0 = blockIdx.y * 16;
  const int b  = blockIdx.z;

  wmma_gemm_tile_64x64(Wo, oatt + b * PLANE, bo, out + b * PLANE,
                       x + b * PLANE, m0, n0, lane);
}

extern "C" void kernel_launch(void* const* d_in, const int* in_sizes, int n_in,
                              void* d_out, int out_size, void* d_ws, size_t ws_size,
                              hipStream_t stream) {
  (void)in_sizes; (void)n_in; (void)out_size; (void)ws_size;

  const float* x  = (const float*)d_in[0];
  const float* Wq = (const float*)d_in[1];
  const float* bq = (const float*)d_in[2];
  const float* Wk = (const float*)d_in[3];
  const float* bk = (const float*)d_in[4];
  const float* Wv = (const float*)d_in[5];
  const float* bv = (const float*)d_in[6];
  const float* Wo = (const float*)d_in[7];
  const float* bo = (const float*)d_in[8];
  float* out = (float*)d_out;

  float* ws   = (float*)d_ws;
  float* q    = ws;                      // BATCH*PLANE = 131072 floats
  float* k    = ws + 1 * BATCH * PLANE;
  float* v    = ws + 2 * BATCH * PLANE;
  float* oatt = ws + 3 * BATCH * PLANE;  // total 2 MB of workspace

  // 1) Fused QKV projections (WMMA f32 GEMM), 1 wave per 16x16 tile.
  qkv_proj_kernel<<<dim3(SPATIAL / 16, DIMS / 16, 3 * BATCH), 32, 0, stream>>>(
      x, Wq, bq, Wk, bk, Wv, bv, q, k, v);

  // 2) Rank-1 softmax attention, one block per head.
  attn_rank1_kernel<<<dim3(BATCH * DIMS), 256, 0, stream>>>(q, k, v, oatt);

  // 3) Output projection + bias + residual.
  out_proj_kernel<<<dim3(SPATIAL / 16, DIMS / 16, BATCH), 32, 0, stream>>>(
      oatt, Wo, bo, x, out);
}